// Quantizer_33036888441549
// MI455X (gfx1250) — compile-verified
//
#include <hip/hip_runtime.h>

typedef __attribute__((ext_vector_type(2))) float v2f;
typedef __attribute__((ext_vector_type(8))) float v8f;

#define NB   65536
#define ND   256
#define NC   2048
#define NWAVES (NB / 16)   // 4096 waves of 16 rows each

// ---------------------------------------------------------------------------
// Kernel 1: code norms ||c||^2, one wave per code row (coalesced).
// ---------------------------------------------------------------------------
__global__ void __launch_bounds__(256) cnorm_kernel(const float* __restrict__ codes,
                                                    float* __restrict__ cn) {
    const int lane = threadIdx.x & 31;
    const int wave = threadIdx.x >> 5;
    const int c    = blockIdx.x * 8 + wave;          // grid = 256 blocks * 8 waves = 2048
    const float* row = codes + (size_t)c * ND;
    float s = 0.0f;
#pragma unroll
    for (int j = 0; j < 8; ++j) {
        float v = row[lane + 32 * j];
        s += v * v;
    }
#pragma unroll
    for (int m = 16; m >= 1; m >>= 1) s += __shfl_xor(s, m, 32);
    if (lane == 0) cn[c] = s;
}

// ---------------------------------------------------------------------------
// Kernel 2: distance GEMM via V_WMMA_F32_16X16X4_F32 + argmin + gather + loss.
// One wave handles 16 rows of x against all 2048 codes.
// ---------------------------------------------------------------------------
__global__ void __launch_bounds__(256, 1) vq_kernel(const float* __restrict__ x,
                                                    const float* __restrict__ codes,
                                                    const float* __restrict__ cn,
                                                    float* __restrict__ outq,
                                                    float* __restrict__ outidx,
                                                    float* __restrict__ losspart) {
    const int lane = threadIdx.x & 31;
    const int lo   = lane & 15;        // M (for A) / N (for B) within tile
    const int hi   = lane >> 4;        // selects K pair {0,1} vs {2,3}
    const int wid  = blockIdx.x * 8 + (threadIdx.x >> 5);
    const int row0 = wid * 16;

    // --- A tile: 16 rows x 256 K kept in registers (64 x v2f = 128 VGPRs) ---
    // ISA 32-bit A 16x4 layout: lane(lo) = M, VGPR j holds K = 2*hi + j.
    v2f a[64];
    const float* xrow = x + (size_t)(row0 + lo) * ND + 2 * hi;
#pragma unroll
    for (int kk = 0; kk < 64; ++kk) {
        a[kk] = *(const v2f*)(xrow + 4 * kk);
    }

    float best[8];
    int   bidx[8];
#pragma unroll
    for (int r = 0; r < 8; ++r) { best[r] = 3.4e38f; bidx[r] = 0; }

    for (int col0 = 0; col0 < NC; col0 += 16) {
        const float* crow = codes + (size_t)(col0 + lo) * ND + 2 * hi;
        if (col0 + 16 < NC) {
            __builtin_prefetch(crow + 16 * ND, 0, 1);   // global_prefetch_b8 next chunk
        }
        v8f acc0 = {}, acc1 = {}, acc2 = {}, acc3 = {};
#pragma unroll
        for (int kk = 0; kk < 64; kk += 4) {
            v2f b0 = *(const v2f*)(crow + 4 * (kk + 0));
            v2f b1 = *(const v2f*)(crow + 4 * (kk + 1));
            v2f b2 = *(const v2f*)(crow + 4 * (kk + 2));
            v2f b3 = *(const v2f*)(crow + 4 * (kk + 3));
            acc0 = __builtin_amdgcn_wmma_f32_16x16x4_f32(false, a[kk + 0], false, b0,
                                                         (short)0, acc0, false, false);
            acc1 = __builtin_amdgcn_wmma_f32_16x16x4_f32(false, a[kk + 1], false, b1,
                                                         (short)0, acc1, false, false);
            acc2 = __builtin_amdgcn_wmma_f32_16x16x4_f32(false, a[kk + 2], false, b2,
                                                         (short)0, acc2, false, false);
            acc3 = __builtin_amdgcn_wmma_f32_16x16x4_f32(false, a[kk + 3], false, b3,
                                                         (short)0, acc3, false, false);
        }
        v8f dot = acc0 + acc1 + acc2 + acc3;

        const float cnv = cn[col0 + lo];   // this lane's column: N = col0 + lo
#pragma unroll
        for (int r = 0; r < 8; ++r) {
            float s = cnv - 2.0f * dot[r];   // ||c||^2 - 2 x.c  (||x||^2 const per row)
            if (s < best[r]) { best[r] = s; bidx[r] = col0 + lo; }
        }
    }

    // --- argmin across the 16 lanes of each half (butterfly; ties -> lower idx) ---
#pragma unroll
    for (int r = 0; r < 8; ++r) {
#pragma unroll
        for (int m = 8; m >= 1; m >>= 1) {
            float ov = __shfl_xor(best[r], m, 32);
            int   oi = __shfl_xor(bidx[r], m, 32);
            if (ov < best[r] || (ov == best[r] && oi < bidx[r])) {
                best[r] = ov; bidx[r] = oi;
            }
        }
    }
    // lanes 0-15 hold rows 0..7 (vgpr r), lanes 16-31 hold rows 8..15
    if (lane == 0) {
#pragma unroll
        for (int r = 0; r < 8; ++r) outidx[row0 + r] = (float)bidx[r];
    }
    if (lane == 16) {
#pragma unroll
        for (int r = 0; r < 8; ++r) outidx[row0 + 8 + r] = (float)bidx[r];
    }

    // --- gather quantized rows + loss partial (coalesced row copies) ---
    float lsum = 0.0f;
#pragma unroll
    for (int m = 0; m < 16; ++m) {
        const int idx = (m < 8) ? __shfl(bidx[m], 0, 32) : __shfl(bidx[m - 8], 16, 32);
        const float* qrow = codes + (size_t)idx * ND;
        const float* xr   = x    + (size_t)(row0 + m) * ND;
        float*       orow = outq + (size_t)(row0 + m) * ND;
#pragma unroll
        for (int t = 0; t < 8; ++t) {
            float q  = qrow[lane + 32 * t];
            float xv = xr[lane + 32 * t];
            orow[lane + 32 * t] = q;
            float d = xv - q;
            lsum += d * d;
        }
    }
#pragma unroll
    for (int m = 16; m >= 1; m >>= 1) lsum += __shfl_xor(lsum, m, 32);
    if (lane == 0) losspart[wid] = lsum;
}

// ---------------------------------------------------------------------------
// Kernel 3: deterministic fixed-order loss reduction (no float atomics).
// loss = mean((x-q)^2) + 0.25*mean((q-x)^2) = 1.25 * mean((x-q)^2)
// ---------------------------------------------------------------------------
__global__ void __launch_bounds__(256) loss_kernel(const float* __restrict__ losspart,
                                                   float* __restrict__ out_loss) {
    __shared__ float sm[256];
    float s = 0.0f;
    for (int i = threadIdx.x; i < NWAVES; i += 256) s += losspart[i];
    sm[threadIdx.x] = s;
    __syncthreads();
    for (int off = 128; off >= 1; off >>= 1) {
        if (threadIdx.x < off) sm[threadIdx.x] += sm[threadIdx.x + off];
        __syncthreads();
    }
    if (threadIdx.x == 0) {
        out_loss[0] = sm[0] * (1.25f / ((float)NB * (float)ND));
    }
}

// ---------------------------------------------------------------------------
extern "C" void kernel_launch(void* const* d_in, const int* in_sizes, int n_in,
                              void* d_out, int out_size, void* d_ws, size_t ws_size,
                              hipStream_t stream) {
    const float* x     = (const float*)d_in[0];   // (65536, 256)
    const float* codes = (const float*)d_in[1];   // (1, 2048, 256)
    // d_in[2] = temperature (unused in eval path)

    float* out     = (float*)d_out;
    float* outq    = out;                               // (65536, 256)
    float* outidx  = out + (size_t)NB * ND;             // (65536,) as float
    float* outloss = outidx + NB;                       // scalar

    float* cn       = (float*)d_ws;                     // 2048 code norms
    float* losspart = cn + NC;                          // 4096 per-wave partials

    cnorm_kernel<<<NC / 8, 256, 0, stream>>>(codes, cn);
    vq_kernel<<<NWAVES / 8, 256, 0, stream>>>(x, codes, cn, outq, outidx, losspart);
    loss_kernel<<<1, 256, 0, stream>>>(losspart, outloss);
}